// ConstraintGNN_75539884802670
// MI455X (gfx1250) — compile-verified
//
#include <hip/hip_runtime.h>
#include <math.h>

#define HID 64

typedef __attribute__((ext_vector_type(2))) float v2f;
typedef __attribute__((ext_vector_type(8))) float v8f;

// ---------------------------------------------------------------- utilities

__global__ void zero_f4_kernel(float4* __restrict__ p, int n4) {
  int i = blockIdx.x * blockDim.x + threadIdx.x;
  if (i < n4) p[i] = make_float4(0.f, 0.f, 0.f, 0.f);
}

// degree count via native integer atomics (accumulated into the dinv buffer)
__global__ void deg_count_kernel(const int* __restrict__ dst, int* __restrict__ cnt, int E) {
  int e = blockIdx.x * blockDim.x + threadIdx.x;
  if (e < E) atomicAdd(&cnt[dst[e]], 1);
}

// in-place int count -> float rsqrt(deg + 1)   (+1 = self loop)
__global__ void make_dinv_kernel(float* buf, int N) {
  int n = blockIdx.x * blockDim.x + threadIdx.x;
  if (n < N) {
    int c = ((const int*)buf)[n];
    buf[n] = rsqrtf((float)c + 1.0f);
  }
}

// layer-1 transform: h1_pre[n][j] = x[n][0]*W1[0][j] + x[n][1]*W1[1][j]
__global__ void h1pre_kernel(const float* __restrict__ x, const float* __restrict__ W1,
                             float* __restrict__ out, int total) {
  int i = blockIdx.x * blockDim.x + threadIdx.x;
  if (i >= total) return;
  int n = i >> 6, j = i & 63;
  out[i] = fmaf(x[2 * n], W1[j], x[2 * n + 1] * W1[64 + j]);
}

// edge scatter-add: 16 threads per edge, float4 gather, 4 native f32 atomics.
// Feature maps (51.2 MB) fit in the 192 MB L2, so this is L2-atomic bound.
__global__ void agg_kernel(const int* __restrict__ src, const int* __restrict__ dst,
                           const float* __restrict__ dinv, const float* __restrict__ H,
                           float* __restrict__ Agg, int E) {
  int gid = blockIdx.x * blockDim.x + threadIdx.x;
  int e = gid >> 4;
  if (e >= E) return;
  int t = (gid & 15) << 2;
  int s = src[e], d = dst[e];
  float norm = dinv[s] * dinv[d];
  float4 h4 = *(const float4*)(H + (size_t)s * HID + t);
  float* o = Agg + (size_t)d * HID + t;
  unsafeAtomicAdd(o + 0, h4.x * norm);   // global_atomic_add_f32
  unsafeAtomicAdd(o + 1, h4.y * norm);
  unsafeAtomicAdd(o + 2, h4.z * norm);
  unsafeAtomicAdd(o + 3, h4.w * norm);
}

// h1[n][j] = relu(agg1 + h1_pre*dinv^2 + b1[j])   (in-place on Agg buffer)
__global__ void post1_kernel(const float* __restrict__ Hpre, float* Agg,
                             const float* __restrict__ dinv, const float* __restrict__ b1,
                             int total) {
  int i = blockIdx.x * blockDim.x + threadIdx.x;
  if (i >= total) return;
  int n = i >> 6, j = i & 63;
  float di = dinv[n];
  float v = Agg[i] + Hpre[i] * di * di + b1[j];
  Agg[i] = fmaxf(v, 0.0f);
}

// h2[n][j] = agg2 + h2_pre*dinv^2 + b2[j]   (no relu; in-place on Hpre buffer)
__global__ void post2_kernel(float* Hpre, const float* __restrict__ Agg,
                             const float* __restrict__ dinv, const float* __restrict__ b2,
                             int total) {
  int i = blockIdx.x * blockDim.x + threadIdx.x;
  if (i >= total) return;
  int n = i >> 6, j = i & 63;
  float di = dinv[n];
  Hpre[i] = Agg[i] + Hpre[i] * di * di + b2[j];
}

// -------------------------------------------------- WMMA fp32 GEMM (K = 64)
//
// One wave computes a 16x16 fp32 tile with 16 chained V_WMMA_F32_16X16X4_F32.
// A 16x4 fp32 fragment (2 VGPR): lanes 0-15 hold K=k,k+1; lanes 16-31 K=k+2,k+3.
// B 4x16 fragment mirrored. C/D: 8 VGPR, M = v + (lane>=16 ? 8 : 0), N = lane&15.
__device__ __forceinline__ v8f wmma_tile_k64(const float* __restrict__ sH, int rowOff,
                                             const float* __restrict__ sW, int col0,
                                             int lane) {
  v8f acc = {};
  int l  = lane & 15;
  int kh = (lane >> 4) << 1;            // 0 for lanes 0-15, 2 for lanes 16-31
  const float* aRow = sH + (rowOff + l) * 65 + kh;
  const float* bCol = sW + kh * HID + col0 + l;
#pragma unroll
  for (int k = 0; k < HID; k += 4) {
    v2f a, b;
    a.x = aRow[k];
    a.y = aRow[k + 1];
    b.x = bCol[k * HID];
    b.y = bCol[(k + 1) * HID];
    acc = __builtin_amdgcn_wmma_f32_16x16x4_f32(false, a, false, b,
                                                (short)0, acc, false, false);
  }
  return acc;
}

// stage W (64x64, 16 KB) and a 32x64 H tile (stride-65 pad vs 64 LDS banks)
__device__ __forceinline__ void stage_tiles(const float* __restrict__ H, int row0g,
                                            const float* __restrict__ W,
                                            float* sW, float* sH, int tid) {
  const float4* W4 = (const float4*)W;
  float4* sW4 = (float4*)sW;
#pragma unroll
  for (int i = tid; i < 1024; i += 256) sW4[i] = W4[i];
#pragma unroll
  for (int i = tid; i < 512; i += 256) {        // 32 rows * 16 float4
    int r = i >> 4, c = (i & 15) << 2;
    float4 v = *(const float4*)(H + (size_t)(row0g + r) * HID + c);
    float* p = sH + r * 65 + c;
    p[0] = v.x; p[1] = v.y; p[2] = v.z; p[3] = v.w;
  }
}

// Out[32 rows x 64 cols per block] = H @ W          (layer-2 transform)
__global__ void __launch_bounds__(256) gemm64_kernel(const float* __restrict__ H,
                                                     const float* __restrict__ W,
                                                     float* __restrict__ Out) {
  __shared__ float sW[HID * HID];
  __shared__ float sH[32 * 65];
  int tid = threadIdx.x;
  int row0g = blockIdx.x * 32;
  stage_tiles(H, row0g, W, sW, sH, tid);
  __syncthreads();

  int wv = tid >> 5, lane = tid & 31;
  int rowOff = (wv >> 2) << 4;                  // 0 or 16
  int col0   = (wv & 3) << 4;                   // 0,16,32,48
  v8f acc = wmma_tile_k64(sH, rowOff, sW, col0, lane);

  int m0 = (lane >> 4) << 3, nn = lane & 15;
  int gr = row0g + rowOff + m0;
#pragma unroll
  for (int v = 0; v < 8; ++v)
    Out[(size_t)(gr + v) * HID + col0 + nn] = acc[v];
}

// out = round(relu(H @ Wfc + bfc))              (final FC, rows < N_VAR)
__global__ void __launch_bounds__(256) fc_kernel(const float* __restrict__ H,
                                                 const float* __restrict__ Wfc,
                                                 const float* __restrict__ bfc,
                                                 float* __restrict__ Out) {
  __shared__ float sW[HID * HID];
  __shared__ float sH[32 * 65];
  int tid = threadIdx.x;
  int row0g = blockIdx.x * 32;
  stage_tiles(H, row0g, Wfc, sW, sH, tid);
  __syncthreads();

  int wv = tid >> 5, lane = tid & 31;
  int rowOff = (wv >> 2) << 4;
  int col0   = (wv & 3) << 4;
  v8f acc = wmma_tile_k64(sH, rowOff, sW, col0, lane);

  int m0 = (lane >> 4) << 3, nn = lane & 15;
  int gr = row0g + rowOff + m0;
  float bias = bfc[col0 + nn];
#pragma unroll
  for (int v = 0; v < 8; ++v) {
    float val = fmaxf(acc[v] + bias, 0.0f);
    Out[(size_t)(gr + v) * HID + col0 + nn] = rintf(val);  // round-half-even == jnp.round
  }
}

// ---------------------------------------------------------------- launcher

extern "C" void kernel_launch(void* const* d_in, const int* in_sizes, int n_in,
                              void* d_out, int out_size, void* d_ws, size_t ws_size,
                              hipStream_t stream) {
  const float* x   = (const float*)d_in[0];
  const int*   ei  = (const int*)  d_in[1];
  const float* W1  = (const float*)d_in[2];
  const float* b1  = (const float*)d_in[3];
  const float* W2  = (const float*)d_in[4];
  const float* b2  = (const float*)d_in[5];
  const float* Wfc = (const float*)d_in[6];
  const float* bfc = (const float*)d_in[7];
  float* out = (float*)d_out;

  const int N    = in_sizes[0] / 2;        // 200000 (divisible by 32)
  const int E    = in_sizes[1] / 2;        // 3200000
  const int B    = in_sizes[7];            // 64
  const int NVAR = out_size / B;           // 112000 (divisible by 32)

  const int* src = ei;
  const int* dst = ei + E;

  // workspace layout: dinv [N] | bufA [N*64] | bufB [N*64]   (~103 MB)
  float* dinv = (float*)d_ws;
  float* bufA = dinv + N;
  float* bufB = bufA + (size_t)N * HID;

  const int T = 256;
  const int nh  = N * HID;                 // 12.8M
  const int nh4 = nh / 4;

  // degrees -> dinv (rsqrt(deg+1)), shared by src and dst normalization
  zero_f4_kernel<<<(N / 4 + T - 1) / T, T, 0, stream>>>((float4*)dinv, N / 4);
  deg_count_kernel<<<(E + T - 1) / T, T, 0, stream>>>(dst, (int*)dinv, E);
  make_dinv_kernel<<<(N + T - 1) / T, T, 0, stream>>>(dinv, N);

  // layer 1: transform, scatter-add, self-loop + bias + relu (in-place in bufB)
  h1pre_kernel<<<(nh + T - 1) / T, T, 0, stream>>>(x, W1, bufA, nh);
  zero_f4_kernel<<<(nh4 + T - 1) / T, T, 0, stream>>>((float4*)bufB, nh4);
  agg_kernel<<<E / 16, T, 0, stream>>>(src, dst, dinv, bufA, bufB, E);
  post1_kernel<<<(nh + T - 1) / T, T, 0, stream>>>(bufA, bufB, dinv, b1, nh);

  // layer 2: fp32-WMMA GEMM (bufB=h1 -> bufA=h2_pre), scatter-add, self+bias
  gemm64_kernel<<<N / 32, T, 0, stream>>>(bufB, W2, bufA);
  zero_f4_kernel<<<(nh4 + T - 1) / T, T, 0, stream>>>((float4*)bufB, nh4);
  agg_kernel<<<E / 16, T, 0, stream>>>(src, dst, dinv, bufA, bufB, E);
  post2_kernel<<<(NVAR * HID + T - 1) / T, T, 0, stream>>>(bufA, bufB, dinv, b2, NVAR * HID);

  // FC on the first N_VAR rows: WMMA GEMM + bias + relu + round
  fc_kernel<<<NVAR / 32, T, 0, stream>>>(bufA, Wfc, bfc, out);
}